// FocalLoss_6674379178398
// MI455X (gfx1250) — compile-verified
//
#include <hip/hip_runtime.h>
#include <hip/hip_bf16.h>

#define ALPHA  0.25f
#define BLOCK  256
#define APT    4            // anchors per thread
#define MAX_ANN_DW 2048     // LDS staging capacity (dwords) for annotations

#if defined(__HIP_DEVICE_COMPILE__) && \
    __has_builtin(__builtin_amdgcn_global_load_async_to_lds_b128) && \
    __has_builtin(__builtin_amdgcn_s_wait_asynccnt)
#define HAVE_ASYNC_LDS 1
#else
#define HAVE_ASYNC_LDS 0
#endif

typedef int   v4i __attribute__((ext_vector_type(4)));
typedef float v4f __attribute__((ext_vector_type(4)));   // native vector: OK for NT builtins

__device__ __forceinline__ float smooth_l1(float d) {
    d = fabsf(d);
    return (d <= (1.0f / 9.0f)) ? 4.5f * d * d : d - (0.5f / 9.0f);
}

__device__ __forceinline__ float clampp(float pc) {
    return fminf(fmaxf(pc, 1e-4f), 1.0f - 1e-4f);
}

// target == 0 element: (1-alpha) * p^2 * (-log(1-p))
__device__ __forceinline__ float neg_term(float pc) {
    pc = clampp(pc);
    return (1.0f - ALPHA) * pc * pc * (-__logf(1.0f - pc));
}

// target == 1 element: alpha * (1-p)^2 * (-log p)
__device__ __forceinline__ float pos_term(float pc) {
    pc = clampp(pc);
    float om = 1.0f - pc;
    return ALPHA * om * om * (-__logf(pc));
}

// Kernel 1: per (anchor-tile, sample) partial sums of cls-loss, reg-loss, num_pos.
__global__ void __launch_bounds__(BLOCK)
focal_partials_kernel(const float* __restrict__ cls,       // [B,A,C]
                      const float* __restrict__ reg,       // [B,A,4]
                      const float* __restrict__ anchors,   // [A,4]
                      const float* __restrict__ ann,       // [B,M,5]
                      float* __restrict__ ws_cls,
                      float* __restrict__ ws_reg,
                      float* __restrict__ ws_pos,
                      int A, int C, int M, int nblk) {
    __shared__ float s_ann[MAX_ANN_DW];
    __shared__ float r_cls[BLOCK / 32], r_reg[BLOCK / 32], r_pos[BLOCK / 32];

    const int b   = blockIdx.y;
    const int blk = blockIdx.x;
    const int tid = threadIdx.x;

    // ---- stage annotations for this sample into LDS ----
    // gfx1250: async global->LDS DMA (b128 chunks), tracked by ASYNCcnt.
    const float* gann = ann + (size_t)b * M * 5;
    const int ndw    = M * 5;                    // 320 dwords for M=64
    const int nchunk = ndw >> 2;                 // full float4 chunks
#if HAVE_ASYNC_LDS
    {
        typedef __attribute__((address_space(1))) v4i* gp_t;   // global (AS1) int4*
        typedef __attribute__((address_space(3))) v4i* lp_t;   // LDS    (AS3) int4*
        for (int i = tid; i < nchunk; i += BLOCK) {
            __builtin_amdgcn_global_load_async_to_lds_b128(
                (gp_t)(uintptr_t)(gann + 4 * i),
                (lp_t)(uintptr_t)(&s_ann[4 * i]),
                0 /*imm offset*/, 0 /*cpol*/);
        }
        __builtin_amdgcn_s_wait_asynccnt(0);
    }
    for (int i = (nchunk << 2) + tid; i < ndw && i < MAX_ANN_DW; i += BLOCK)
        s_ann[i] = gann[i];                      // tail dwords (none when M%4==0)
#else
    for (int i = tid; i < ndw && i < MAX_ANN_DW; i += BLOCK) s_ann[i] = gann[i];
#endif
    __syncthreads();

    // ---- per-thread anchor state (4 anchors, consecutive-lane coalesced) ----
    const int a_base = blk * (BLOCK * APT) + tid;
    v4f  ab[APT];
    float best[APT];
    int   bm[APT];
    bool  valid[APT];
#pragma unroll
    for (int it = 0; it < APT; ++it) {
        const int a = a_base + it * BLOCK;
        valid[it]   = (a < A);
        if (valid[it]) {
            ab[it] = ((const v4f*)anchors)[a];
        } else {
            ab[it].x = 0.0f; ab[it].y = 0.0f; ab[it].z = 1.0f; ab[it].w = 1.0f;
        }
        best[it] = -1.0f;
        bm[it]   = 0;
    }

    // ---- IoU max/argmax: m outer, anchors inner (4x fewer LDS broadcasts) ----
    for (int m = 0; m < M; ++m) {
        const float bx1 = s_ann[m * 5 + 0];
        const float by1 = s_ann[m * 5 + 1];
        const float bx2 = s_ann[m * 5 + 2];
        const float by2 = s_ann[m * 5 + 3];
        const float areaB = (bx2 - bx1) * (by2 - by1);
#pragma unroll
        for (int it = 0; it < APT; ++it) {
            float iw = fminf(ab[it].z, bx2) - fmaxf(ab[it].x, bx1); iw = fmaxf(iw, 0.0f);
            float ih = fminf(ab[it].w, by2) - fmaxf(ab[it].y, by1); ih = fmaxf(ih, 0.0f);
            const float inter = iw * ih;
            const float areaA = (ab[it].z - ab[it].x) * (ab[it].w - ab[it].y);
            const float ua  = fmaxf(areaA + areaB - inter, 1e-8f);
            const float iou = inter / ua;
            if (iou > best[it]) { best[it] = iou; bm[it] = m; }   // first-max argmax
        }
    }

    float cls_acc = 0.0f, reg_acc = 0.0f, pos_acc = 0.0f;

#pragma unroll
    for (int it = 0; it < APT; ++it) {
        if (!valid[it]) continue;
        const int a    = a_base + it * BLOCK;
        const bool pos = best[it] >= 0.5f;
        const bool neg = best[it] <  0.4f;

        // ---- classification focal loss (ignored anchors read zero bytes) ----
        if (pos || neg) {
            const float* row = cls + ((size_t)b * A + a) * C;
            const v4f*   c4  = (const v4f*)row;
            const int nv = C >> 2;
            float acc = 0.0f;
            for (int j = 0; j < nv; ++j) {
                const v4f p = __builtin_nontemporal_load(c4 + j);   // streamed once
                acc += neg_term(p.x) + neg_term(p.y) + neg_term(p.z) + neg_term(p.w);
            }
            for (int c = nv << 2; c < C; ++c)                 // tail (C%4 != 0)
                acc += neg_term(row[c]);
            if (pos) {                                        // class-0 correction
                const float p0 = row[0];                      // L0 hit (just loaded)
                acc += pos_term(p0) - neg_term(p0);
            }
            cls_acc += acc;
        }

        // ---- regression smooth-L1 for positives ----
        if (pos) {
            const float aw  = ab[it].z - ab[it].x;
            const float ah  = ab[it].w - ab[it].y;
            const float acx = ab[it].x + 0.5f * aw;
            const float acy = ab[it].y + 0.5f * ah;
            const float gx1 = s_ann[bm[it] * 5 + 0];
            const float gy1 = s_ann[bm[it] * 5 + 1];
            const float gx2 = s_ann[bm[it] * 5 + 2];
            const float gy2 = s_ann[bm[it] * 5 + 3];
            float gw = gx2 - gx1, gh = gy2 - gy1;
            const float gcx = gx1 + 0.5f * gw;
            const float gcy = gy1 + 0.5f * gh;
            gw = fmaxf(gw, 1.0f); gh = fmaxf(gh, 1.0f);
            const float t0 = (gcx - acx) / aw * 10.0f;        // / std 0.1
            const float t1 = (gcy - acy) / ah * 10.0f;
            const float t2 = __logf(gw / aw) * 5.0f;          // / std 0.2
            const float t3 = __logf(gh / ah) * 5.0f;
            const v4f rg = __builtin_nontemporal_load(
                (const v4f*)(reg + ((size_t)b * A + a) * 4));
            reg_acc += smooth_l1(t0 - rg.x) + smooth_l1(t1 - rg.y) +
                       smooth_l1(t2 - rg.z) + smooth_l1(t3 - rg.w);
            pos_acc += 1.0f;
        }
    }

    // ---- deterministic block reduction: wave32 shuffles + 8-wave LDS tree ----
    const unsigned lane = tid & 31u, wid = tid >> 5;
#pragma unroll
    for (int off = 16; off > 0; off >>= 1) {
        cls_acc += __shfl_down(cls_acc, off, 32);
        reg_acc += __shfl_down(reg_acc, off, 32);
        pos_acc += __shfl_down(pos_acc, off, 32);
    }
    if (lane == 0) { r_cls[wid] = cls_acc; r_reg[wid] = reg_acc; r_pos[wid] = pos_acc; }
    __syncthreads();
    if (tid == 0) {
        float c = 0.0f, r = 0.0f, p = 0.0f;
#pragma unroll
        for (int i = 0; i < BLOCK / 32; ++i) { c += r_cls[i]; r += r_reg[i]; p += r_pos[i]; }
        ws_cls[(size_t)b * nblk + blk] = c;
        ws_reg[(size_t)b * nblk + blk] = r;
        ws_pos[(size_t)b * nblk + blk] = p;
    }
}

// Kernel 2: one wave per sample reduces partials (fixed order -> deterministic),
// applies per-sample normalization, thread 0 writes the two batch means.
__global__ void __launch_bounds__(256)
focal_finalize_kernel(const float* __restrict__ ws_cls,
                      const float* __restrict__ ws_reg,
                      const float* __restrict__ ws_pos,
                      float* __restrict__ out, int B, int nblk) {
    __shared__ float s_cl[64], s_rl[64];
    const int lane = threadIdx.x & 31, wid = threadIdx.x >> 5, nw = blockDim.x >> 5;
    for (int b = wid; b < B; b += nw) {
        float c = 0.0f, r = 0.0f, p = 0.0f;
        for (int i = lane; i < nblk; i += 32) {
            c += ws_cls[(size_t)b * nblk + i];
            r += ws_reg[(size_t)b * nblk + i];
            p += ws_pos[(size_t)b * nblk + i];
        }
#pragma unroll
        for (int off = 16; off > 0; off >>= 1) {
            c += __shfl_down(c, off, 32);
            r += __shfl_down(r, off, 32);
            p += __shfl_down(p, off, 32);
        }
        if (lane == 0 && b < 64) {
            s_cl[b] = c / fmaxf(p, 1.0f);
            s_rl[b] = (p > 0.0f) ? r / fmaxf(p * 4.0f, 1.0f) : 0.0f;
        }
    }
    __syncthreads();
    if (threadIdx.x == 0) {
        float mc = 0.0f, mr = 0.0f;
        for (int b = 0; b < B && b < 64; ++b) { mc += s_cl[b]; mr += s_rl[b]; }
        out[0] = mc / (float)B;
        out[1] = mr / (float)B;
    }
}

extern "C" void kernel_launch(void* const* d_in, const int* in_sizes, int n_in,
                              void* d_out, int out_size, void* d_ws, size_t ws_size,
                              hipStream_t stream) {
    const float* cls     = (const float*)d_in[0];   // [B,A,C]
    const float* reg     = (const float*)d_in[1];   // [B,A,4]
    const float* anchors = (const float*)d_in[2];   // [1,A,4]
    const float* ann     = (const float*)d_in[3];   // [B,M,5]

    const int A = in_sizes[2] / 4;
    const int B = in_sizes[1] / (4 * A);
    const int M = in_sizes[3] / (5 * B);
    const int C = (int)((long long)in_sizes[0] / ((long long)B * A));

    const int per_blk = BLOCK * APT;
    const int nblk = (A + per_blk - 1) / per_blk;

    float* ws     = (float*)d_ws;
    float* ws_cls = ws;
    float* ws_reg = ws + (size_t)B * nblk;
    float* ws_pos = ws + 2 * (size_t)B * nblk;

    dim3 grid(nblk, B);
    focal_partials_kernel<<<grid, BLOCK, 0, stream>>>(cls, reg, anchors, ann,
                                                      ws_cls, ws_reg, ws_pos,
                                                      A, C, M, nblk);
    focal_finalize_kernel<<<1, 256, 0, stream>>>(ws_cls, ws_reg, ws_pos,
                                                 (float*)d_out, B, nblk);
}